// Multimodal_LDM_19447611916775
// MI455X (gfx1250) — compile-verified
//
#include <hip/hip_runtime.h>
#include <hip/hip_bf16.h>
#include <math.h>

typedef __attribute__((ext_vector_type(2))) float v2f;
typedef __attribute__((ext_vector_type(8))) float v8f;

#define N_PP   8192
#define N_A    8192
#define DIM    16
#define E_CNT  262144
#define TILES_PER_DIM 512                 // 8192 / 16
#define NL_BLOCKS_PER_MAT 4096            // 4096 blocks * 8 waves * 8 tiles = 262144
#define NL_WAVES_PER_MAT  (NL_BLOCKS_PER_MAT * 8)       // 32768

#define LOG2E 1.4426950408889634f

// fast hardware paths: single v_sqrt_f32 / v_exp_f32 (adequate vs jnp f32 ref)
__device__ __forceinline__ float fast_sqrt(float x) {
    return __builtin_amdgcn_sqrtf(x);
}
__device__ __forceinline__ float fast_exp(float x) {
    return __builtin_amdgcn_exp2f(x * LOG2E);
}

// ---------------------------------------------------------------------------
// Kernel 1: per-row sum of squares for p_star, p, a -> contiguous sq[3*8192]
// ---------------------------------------------------------------------------
__global__ __launch_bounds__(256) void prep_sumsq_kernel(
    const float* __restrict__ ps, const float* __restrict__ p,
    const float* __restrict__ a, float* __restrict__ sq)
{
    int i = blockIdx.x * 256 + threadIdx.x;
    if (i >= 3 * N_PP) return;
    int which = i >> 13;            // 0=p_star, 1=p, 2=a
    int r     = i & (N_PP - 1);
    const float* src = (which == 0) ? ps : ((which == 1) ? p : a);
    const float* row = src + (size_t)r * DIM;
    float s = 0.f;
#pragma unroll
    for (int k = 0; k < DIM; k += 4) {
        float4 v = *(const float4*)(row + k);
        s += v.x * v.x + v.y * v.y + v.z * v.z + v.w * v.w;
    }
    sq[i] = s;
}

// ---------------------------------------------------------------------------
// Kernel 2: dense nonlink terms via f32 WMMA tiles.
//   blockIdx.x in [0, 4096)      -> pp matrix (strict upper triangle)
//   blockIdx.x in [4096, 8192)   -> ap matrix (full)
// One wave owns one 16x16 tile per iteration; 4x V_WMMA_F32_16X16X4_F32
// accumulate -2 * X @ Y^T for the tile (K = 16 split into 4 chunks of 4).
// ---------------------------------------------------------------------------
__global__ __launch_bounds__(256) void nonlink_wmma_kernel(
    const float* __restrict__ ps, const float* __restrict__ p,
    const float* __restrict__ a,
    const float* __restrict__ gamma_pp, const float* __restrict__ beta_ap,
    const float* __restrict__ sq,            // [sq_ps | sq_p | sq_a]
    float* __restrict__ partial)             // one partial per block (8192)
{
    const int tid  = threadIdx.x;
    const int lane = tid & 31;
    const int wave = tid >> 5;
    const bool isAP = (blockIdx.x >= NL_BLOCKS_PER_MAT);
    const int  blk  = isAP ? (blockIdx.x - NL_BLOCKS_PER_MAT) : blockIdx.x;

    const float* Y   = isAP ? a : p;
    const float* sqX = sq;                                   // sq_ps
    const float* sqY = isAP ? (sq + 2 * N_PP) : (sq + N_PP);
    const float* cfX = isAP ? beta_ap : gamma_pp;            // coef for row node m
    const float* cfY = isAP ? (beta_ap + N_PP) : (gamma_pp + N_PP); // coef for col n

    const int half = lane >> 4;      // 0: K pair {0,1}; 1: K pair {2,3} per chunk
    const int lr   = lane & 15;

    float acc = 0.f;
    const int wgid = blk * 8 + wave;                 // 0 .. 32767 (per matrix)
#pragma unroll 1
    for (int s = 0; s < 8; ++s) {
        const int t  = wgid + s * NL_WAVES_PER_MAT;  // 0 .. 262143
        const int m0 = (t >> 9) << 4;                // tile row base (p_star index)
        const int n0 = (t & (TILES_PER_DIM - 1)) << 4; // tile col base
        // pp: strict upper triangle (i < j); skip tiles entirely on/below diag.
        // Condition is wave-uniform -> EXEC stays all ones for the WMMAs.
        if (!isAP && m0 >= n0 + 16) continue;

        const float* xrow = ps + (size_t)(m0 + lr) * DIM + half * 2;
        const float* yrow = Y  + (size_t)(n0 + lr) * DIM + half * 2;

        v8f c = {};
#pragma unroll
        for (int kb = 0; kb < DIM; kb += 4) {
            v2f af, bf;
            af[0] = -2.0f * xrow[kb];
            af[1] = -2.0f * xrow[kb + 1];
            bf[0] = yrow[kb];
            bf[1] = yrow[kb + 1];
            // D = A(16x4) * B(4x16) + C, f32 in / f32 out
            c = __builtin_amdgcn_wmma_f32_16x16x4_f32(
                    false, af, false, bf, (short)0, c, false, false);
        }

        const int   n  = n0 + lr;
        const float y2 = sqY[n];
        const float gy = cfY[n];
#pragma unroll
        for (int r = 0; r < 8; ++r) {
            const int m  = m0 + half * 8 + r;        // C/D layout: VGPR r -> M=r(+8)
            float d2 = sqX[m] + y2 + c[r];           // c already holds -2*x.y
            float d  = fast_sqrt(fmaxf(d2, 0.f));
            float v  = fast_exp(cfX[m] + gy - d);
            bool masked = (!isAP) && (m >= n);       // keep strict upper for pp
            acc += masked ? 0.f : v;
        }
    }

    __shared__ float red[256];
    red[tid] = acc;
    __syncthreads();
#pragma unroll
    for (int off = 128; off > 0; off >>= 1) {
        if (tid < off) red[tid] += red[tid + off];
        __syncthreads();
    }
    if (tid == 0) partial[blockIdx.x] = red[0];
}

// ---------------------------------------------------------------------------
// Kernel 3: link terms over edges.
//   blockIdx.x in [0,1024)    -> pp edges
//   blockIdx.x in [1024,2048) -> ap edges
// ---------------------------------------------------------------------------
__global__ __launch_bounds__(256) void edge_kernel(
    const float* __restrict__ ps, const float* __restrict__ p,
    const float* __restrict__ a,
    const float* __restrict__ gamma_pp, const float* __restrict__ beta_ap,
    const int* __restrict__ edges_pp, const int* __restrict__ edges_ap,
    float* __restrict__ partial)             // one partial per block (2048)
{
    const int tid  = threadIdx.x;
    const bool isAP = (blockIdx.x >= 1024);
    const int  blk  = isAP ? (blockIdx.x - 1024) : blockIdx.x;
    const int  idx  = blk * 256 + tid;       // 0 .. E-1

    const int* e = isAP ? edges_ap : edges_pp;
    const int i0 = e[idx];                   // row 0 of (2,E)
    const int i1 = e[idx + E_CNT];           // row 1 of (2,E)

    const float* x = ps + (size_t)i0 * DIM;
    const float* y = isAP ? (a + (size_t)(i1 - N_PP) * DIM)
                          : (p + (size_t)i1 * DIM);
    const float cf = isAP ? (beta_ap[i0] + beta_ap[i1])
                          : (gamma_pp[i0] + gamma_pp[i1 + N_PP]);

    float d2 = 0.f;
#pragma unroll
    for (int k = 0; k < DIM; k += 4) {
        float4 xv = *(const float4*)(x + k);
        float4 yv = *(const float4*)(y + k);
        float dx = xv.x - yv.x, dy = xv.y - yv.y;
        float dz = xv.z - yv.z, dw = xv.w - yv.w;
        d2 += dx * dx + dy * dy + dz * dz + dw * dw;
    }
    float val = cf - fast_sqrt(d2);

    __shared__ float red[256];
    red[tid] = val;
    __syncthreads();
#pragma unroll
    for (int off = 128; off > 0; off >>= 1) {
        if (tid < off) red[tid] += red[tid + off];
        __syncthreads();
    }
    if (tid == 0) partial[blockIdx.x] = red[0];
}

// ---------------------------------------------------------------------------
// Kernel 4: deterministic final reduction (double accumulation) -> scalar out
// ---------------------------------------------------------------------------
__global__ __launch_bounds__(256) void final_kernel(
    const float* __restrict__ pnl,   // 8192: [pp nonlink | ap nonlink]
    const float* __restrict__ peg,   // 2048: [pp link | ap link]
    float* __restrict__ out)
{
    __shared__ double red[256];
    const int tid = threadIdx.x;
    double res[4];
#pragma unroll 1
    for (int q = 0; q < 4; ++q) {
        const float* src; int n;
        if      (q == 0) { src = pnl;                     n = NL_BLOCKS_PER_MAT; }
        else if (q == 1) { src = pnl + NL_BLOCKS_PER_MAT; n = NL_BLOCKS_PER_MAT; }
        else if (q == 2) { src = peg;                     n = 1024; }
        else             { src = peg + 1024;              n = 1024; }
        double acc = 0.0;
        for (int i = tid; i < n; i += 256) acc += (double)src[i];
        red[tid] = acc;
        __syncthreads();
#pragma unroll
        for (int off = 128; off > 0; off >>= 1) {
            if (tid < off) red[tid] += red[tid + off];
            __syncthreads();
        }
        res[q] = red[0];
        __syncthreads();
    }
    if (tid == 0) {
        double nonlink_pp = res[0], nonlink_ap = res[1];
        double link_pp    = res[2], link_ap    = res[3];
        double nll_pp = -(link_pp - nonlink_pp);
        double nll_ap = -(link_ap - nonlink_ap);
        out[0] = (float)(0.5 * nll_pp / (double)N_PP +
                         0.5 * nll_ap / (double)N_A);
    }
}

// ---------------------------------------------------------------------------
// Launcher
// ---------------------------------------------------------------------------
extern "C" void kernel_launch(void* const* d_in, const int* in_sizes, int n_in,
                              void* d_out, int out_size, void* d_ws, size_t ws_size,
                              hipStream_t stream) {
    const float* p_star   = (const float*)d_in[0];   // (8192,16)
    const float* p        = (const float*)d_in[1];   // (8192,16)
    const float* a        = (const float*)d_in[2];   // (8192,16)
    const float* beta_ap  = (const float*)d_in[3];   // (16384,)
    const float* gamma_pp = (const float*)d_in[4];   // (16384,)
    const int*   edges_pp = (const int*)d_in[5];     // (2,E) row-major
    const int*   edges_ap = (const int*)d_in[6];     // (2,E) row-major
    float* out = (float*)d_out;

    // workspace layout (floats):
    //   [0,8192)        nonlink partials (pp | ap)
    //   [8192,10240)    edge partials (pp | ap)
    //   [10240,34816)   sumsq rows: sq_ps | sq_p | sq_a
    float* ws         = (float*)d_ws;
    float* partial_nl = ws;
    float* partial_eg = ws + 8192;
    float* sq         = ws + 10240;

    prep_sumsq_kernel<<<(3 * N_PP + 255) / 256, 256, 0, stream>>>(
        p_star, p, a, sq);

    nonlink_wmma_kernel<<<2 * NL_BLOCKS_PER_MAT, 256, 0, stream>>>(
        p_star, p, a, gamma_pp, beta_ap, sq, partial_nl);

    edge_kernel<<<2048, 256, 0, stream>>>(
        p_star, p, a, gamma_pp, beta_ap, edges_pp, edges_ap, partial_eg);

    final_kernel<<<1, 256, 0, stream>>>(partial_nl, partial_eg, out);
}